// Mamba2_45913200394565
// MI455X (gfx1250) — compile-verified
//
#include <hip/hip_runtime.h>
#include <hip/hip_bf16.h>

// Mamba2 (SSD) forward for gfx1250. All matmuls use V_WMMA_F32_16X16X4_F32
// (exact fp32 semantics on the matrix pipe, wave32 layout).

#define L_SEQ   4096
#define DMODEL  2048
#define DINNER  4096
#define DSTATE  128
#define NH      64
#define HD      64
#define NPROJ   8512   // 2*DINNER + 2*DSTATE + NH
#define CONVDIM 4352   // DINNER + 2*DSTATE
#define QCH     256
#define NCHUNK  16

typedef __attribute__((ext_vector_type(2))) float v2f;
typedef __attribute__((ext_vector_type(8))) float v8f;

static __device__ __forceinline__ v8f wmma4(v2f a, v2f b, v8f c) {
  // D = A(16x4) * B(4x16) + C(16x16), fp32
  return __builtin_amdgcn_wmma_f32_16x16x4_f32(false, a, false, b, (short)0, c,
                                               false, false);
}

// A-style fragment load from a row-major matrix: element(row0+lane%16, k0+2*(lane/16)+r)
static __device__ __forceinline__ v2f load_frag_rm(const float* p, int ld, int row0, int k0) {
  int lane = threadIdx.x & 31;
  const float* q = p + (size_t)(row0 + (lane & 15)) * ld + k0 + ((lane >> 4) << 1);
  v2f v; v.x = q[0]; v.y = q[1];
  return v;
}

// C/D tile store: VGPR i holds row m0 + i + 8*(lane/16), col n0 + lane%16
static __device__ __forceinline__ void store_tile(float* p, int ld, int m0, int n0, v8f c) {
  int lane = threadIdx.x & 31;
  int col = n0 + (lane & 15);
  int rb  = m0 + ((lane >> 4) << 3);
#pragma unroll
  for (int i = 0; i < 8; ++i) p[(size_t)(rb + i) * ld + col] = c[i];
}

// ---------------------------------------------------------------------------
// Generic batched C = A(MxK) * B(NxK)^T, all dims multiples of 64.
// One wave computes a 64x64 tile (4x4 accumulators). 8 waves per block.
// ---------------------------------------------------------------------------
__global__ __launch_bounds__(256) void gemm_abt_wmma(
    const float* __restrict__ A, const float* __restrict__ B, float* __restrict__ C,
    int M, int N, int K, long long sA, long long sB, long long sC) {
  const float* Ab = A + (long long)blockIdx.z * sA;
  const float* Bb = B + (long long)blockIdx.z * sB;
  float*       Cb = C + (long long)blockIdx.z * sC;
  int ntn = N >> 6, ntm = M >> 6;
  int wave = threadIdx.x >> 5;
  int tile = blockIdx.x * 8 + wave;
  if (tile >= ntm * ntn) return;          // wave-uniform: EXEC stays full
  int m0 = (tile / ntn) << 6;
  int n0 = (tile % ntn) << 6;

  v8f acc[4][4] = {};
  for (int k0 = 0; k0 < K; k0 += 4) {
    v2f a[4], b[4];
#pragma unroll
    for (int i = 0; i < 4; ++i) a[i] = load_frag_rm(Ab, K, m0 + 16 * i, k0);
#pragma unroll
    for (int j = 0; j < 4; ++j) b[j] = load_frag_rm(Bb, K, n0 + 16 * j, k0);
#pragma unroll
    for (int i = 0; i < 4; ++i)
#pragma unroll
      for (int j = 0; j < 4; ++j) acc[i][j] = wmma4(a[i], b[j], acc[i][j]);
  }
#pragma unroll
  for (int i = 0; i < 4; ++i)
#pragma unroll
    for (int j = 0; j < 4; ++j) store_tile(Cb, N, m0 + 16 * i, n0 + 16 * j, acc[i][j]);
}

// ---------------------------------------------------------------------------
// dt = softplus(raw + bias), dA = dt * (-exp(A_log));  L*NH threads
// ---------------------------------------------------------------------------
__global__ __launch_bounds__(256) void dt_da_kernel(
    const float* __restrict__ zxbcdt, const float* __restrict__ dt_bias,
    const float* __restrict__ A_log, float* __restrict__ dt_o, float* __restrict__ dA_o) {
  int idx = blockIdx.x * 256 + threadIdx.x;   // t*64 + h
  int h = idx & 63, t = idx >> 6;
  float raw = zxbcdt[(size_t)t * NPROJ + (NPROJ - NH) + h] + dt_bias[h];
  float dt = (raw > 20.f) ? raw : log1pf(__expf(raw));
  dt_o[idx] = dt;
  dA_o[idx] = dt * (-__expf(A_log[h]));
}

// ---------------------------------------------------------------------------
// Causal depthwise conv(4) + SiLU; splits into x / Xdt / B / C.  L*CONVDIM threads
// ---------------------------------------------------------------------------
__global__ __launch_bounds__(256) void conv_silu_kernel(
    const float* __restrict__ zxbcdt, const float* __restrict__ conv_w,
    const float* __restrict__ conv_b, const float* __restrict__ dt,
    float* __restrict__ x, float* __restrict__ Xdt,
    float* __restrict__ Bm, float* __restrict__ Cm) {
  long long idx = (long long)blockIdx.x * 256 + threadIdx.x;   // t*CONVDIM + j
  int j = (int)(idx % CONVDIM);
  int t = (int)(idx / CONVDIM);
  const float* col = zxbcdt + DINNER + j;      // xBC column, row stride NPROJ
  float acc = conv_b[j];
#pragma unroll
  for (int k = 0; k < 4; ++k) {
    int tt = t - 3 + k;
    float v = (tt >= 0) ? col[(size_t)tt * NPROJ] : 0.f;
    acc += v * conv_w[j * 4 + k];
  }
  float s = acc / (1.f + __expf(-acc));        // SiLU
  if (j < DINNER) {
    x[(size_t)t * DINNER + j]   = s;
    Xdt[(size_t)t * DINNER + j] = s * dt[t * NH + (j >> 6)];
  } else if (j < DINNER + DSTATE) {
    Bm[(size_t)t * DSTATE + (j - DINNER)] = s;
  } else {
    Cm[(size_t)t * DSTATE + (j - DINNER - DSTATE)] = s;
  }
}

// ---------------------------------------------------------------------------
// Per-(h,c) inclusive cumsum of dA over the chunk; also emits chunk_sum.
// ---------------------------------------------------------------------------
__global__ __launch_bounds__(256) void cumsum_kernel(
    const float* __restrict__ dA, float* __restrict__ Acum, float* __restrict__ chunk_sum) {
  int h = blockIdx.x >> 4, c = blockIdx.x & 15;
  int q = threadIdx.x;
  __shared__ float buf[QCH];
  buf[q] = dA[(size_t)(c * QCH + q) * NH + h];
  __syncthreads();
  for (int off = 1; off < QCH; off <<= 1) {
    float t = (q >= off) ? buf[q - off] : 0.f;
    __syncthreads();
    buf[q] += t;
    __syncthreads();
  }
  Acum[((size_t)h * NCHUNK + c) * QCH + q] = buf[q];
  if (q == QCH - 1) chunk_sum[h * NCHUNK + c] = buf[q];
}

// ---------------------------------------------------------------------------
// states[c,h,p,n] = sum_l B[c,l,n] * exp(cs - Acum[l]) * Xdt[c,l,h,p]
// One block per (c,h); wave w owns n-tile w, 4 p-tiles; K = 256.
// ---------------------------------------------------------------------------
__global__ __launch_bounds__(256) void states_kernel(
    const float* __restrict__ Xdt, const float* __restrict__ Bm,
    const float* __restrict__ Acum, const float* __restrict__ chunk_sum,
    float* __restrict__ states) {
  int c = blockIdx.x & 15, h = blockIdx.x >> 4;
  int tid = threadIdx.x, wave = tid >> 5, lane = tid & 31;
  __shared__ float w[QCH];
  float cs = chunk_sum[h * NCHUNK + c];
  w[tid] = __expf(cs - Acum[((size_t)h * NCHUNK + c) * QCH + tid]);
  __syncthreads();

  const float* Xb = Xdt + (size_t)(c * QCH) * DINNER + h * HD;
  const float* Bb = Bm + (size_t)(c * QCH) * DSTATE;
  int n0 = wave * 16;
  v8f acc[4] = {};
  for (int k0 = 0; k0 < QCH; k0 += 4) {
    int l = k0 + ((lane >> 4) << 1);
    v2f bf;
    bf.x = Bb[(size_t)l * DSTATE + n0 + (lane & 15)];
    bf.y = Bb[(size_t)(l + 1) * DSTATE + n0 + (lane & 15)];
    float w0 = w[l], w1 = w[l + 1];
#pragma unroll
    for (int i = 0; i < 4; ++i) {
      int p = i * 16 + (lane & 15);
      v2f af;
      af.x = Xb[(size_t)l * DINNER + p] * w0;
      af.y = Xb[(size_t)(l + 1) * DINNER + p] * w1;
      acc[i] = wmma4(af, bf, acc[i]);
    }
  }
  float* S = states + ((size_t)(c * NH + h) * HD) * DSTATE;
#pragma unroll
  for (int i = 0; i < 4; ++i) store_tile(S, DSTATE, i * 16, n0, acc[i]);
}

// ---------------------------------------------------------------------------
// 16-step inter-chunk recurrence: prev[c] = states[c-1] + exp(chunk_sum[c-1])*prev[c-1]
// One thread per (h,p,n).
// ---------------------------------------------------------------------------
__global__ __launch_bounds__(256) void chunkscan_kernel(
    const float* __restrict__ states, const float* __restrict__ chunk_sum,
    float* __restrict__ prev_states) {
  int idx = blockIdx.x * 256 + threadIdx.x;   // h*HD*DSTATE + p*DSTATE + n
  int n = idx & (DSTATE - 1);
  int p = (idx >> 7) & (HD - 1);
  int h = idx >> 13;
  float s = 0.f;
  for (int c = 0; c < NCHUNK; ++c) {
    size_t o = ((size_t)(c * NH + h) * HD + p) * DSTATE + n;
    prev_states[o] = s;
    s = states[o] + __expf(chunk_sum[h * NCHUNK + c]) * s;
  }
}

// ---------------------------------------------------------------------------
// Y[c,l,h,p] = sum_{s<=l} exp(Acum[l]-Acum[s])*scores[l,s]*Xdt[c,s,h,p]
//            + exp(Acum[l]) * sum_n C[c,l,n]*prev_states[c,h,p,n]
//            + D[h]*x[c,l,h,p]
// One block per (c,h); wave w owns l-tiles 2w..2w+1 x all 4 p-tiles.
// Both GEMMs accumulate into the same WMMA accumulators.
// ---------------------------------------------------------------------------
__global__ __launch_bounds__(256) void y_kernel(
    const float* __restrict__ Xdt, const float* __restrict__ Cm,
    const float* __restrict__ scores, const float* __restrict__ Acum,
    const float* __restrict__ prev_states, const float* __restrict__ x,
    const float* __restrict__ D_param, float* __restrict__ y) {
  int c = blockIdx.x & 15, h = blockIdx.x >> 4;
  int tid = threadIdx.x, wave = tid >> 5, lane = tid & 31;
  __shared__ float ac[QCH];
  __shared__ float eA[QCH];
  float a = Acum[((size_t)h * NCHUNK + c) * QCH + tid];
  ac[tid] = a;
  eA[tid] = __expf(a);
  __syncthreads();

  int l0 = wave * 32;
  v8f acc[2][4] = {};
  const float* Sc = scores + (size_t)c * QCH * QCH;
  const float* Xb = Xdt + (size_t)(c * QCH) * DINNER + h * HD;

  // Phase 1: masked-decay diagonal block (triangular K loop over s)
  for (int s0 = 0; s0 <= l0 + 28; s0 += 4) {
    int s = s0 + ((lane >> 4) << 1);
    float as0 = ac[s], as1 = ac[s + 1];
    v2f bf[4];
#pragma unroll
    for (int j = 0; j < 4; ++j) {
      int p = j * 16 + (lane & 15);
      bf[j].x = Xb[(size_t)s * DINNER + p];
      bf[j].y = Xb[(size_t)(s + 1) * DINNER + p];
    }
#pragma unroll
    for (int i = 0; i < 2; ++i) {
      int l = l0 + i * 16 + (lane & 15);
      float al = ac[l];
      v2f af;
      af.x = (l >= s)     ? __expf(al - as0) * Sc[(size_t)l * QCH + s]     : 0.f;
      af.y = (l >= s + 1) ? __expf(al - as1) * Sc[(size_t)l * QCH + s + 1] : 0.f;
#pragma unroll
      for (int j = 0; j < 4; ++j) acc[i][j] = wmma4(af, bf[j], acc[i][j]);
    }
  }

  // Phase 2: inter-chunk contribution (K over state dim n)
  const float* Cb = Cm + (size_t)(c * QCH) * DSTATE;
  const float* PS = prev_states + (size_t)(c * NH + h) * HD * DSTATE;
  for (int n0 = 0; n0 < DSTATE; n0 += 4) {
    int n = n0 + ((lane >> 4) << 1);
    v2f bf[4];
#pragma unroll
    for (int j = 0; j < 4; ++j) {
      int p = j * 16 + (lane & 15);
      bf[j].x = PS[(size_t)p * DSTATE + n];
      bf[j].y = PS[(size_t)p * DSTATE + n + 1];
    }
#pragma unroll
    for (int i = 0; i < 2; ++i) {
      int l = l0 + i * 16 + (lane & 15);
      float e = eA[l];
      v2f af;
      af.x = Cb[(size_t)l * DSTATE + n] * e;
      af.y = Cb[(size_t)l * DSTATE + n + 1] * e;
#pragma unroll
      for (int j = 0; j < 4; ++j) acc[i][j] = wmma4(af, bf[j], acc[i][j]);
    }
  }

  // Phase 3: skip term + store
  float Dp = D_param[h];
  const float* xb = x + (size_t)(c * QCH) * DINNER + h * HD;
  float* yb = y + (size_t)(c * QCH) * DINNER + h * HD;
#pragma unroll
  for (int i = 0; i < 2; ++i) {
    int lb = l0 + i * 16 + ((lane >> 4) << 3);
#pragma unroll
    for (int j = 0; j < 4; ++j) {
      int p = j * 16 + (lane & 15);
#pragma unroll
      for (int r = 0; r < 8; ++r) {
        int l = lb + r;
        yb[(size_t)l * DINNER + p] = acc[i][j][r] + Dp * xb[(size_t)l * DINNER + p];
      }
    }
  }
}

// ---------------------------------------------------------------------------
// y = y * silu(z); RMSNorm over DINNER with norm_w. One block per row t.
// ---------------------------------------------------------------------------
__global__ __launch_bounds__(256) void gate_norm_kernel(
    const float* __restrict__ y, const float* __restrict__ zxbcdt,
    const float* __restrict__ norm_w, float* __restrict__ yn) {
  int t = blockIdx.x;
  __shared__ float red[256];
  float ss = 0.f;
  for (int d = threadIdx.x; d < DINNER; d += 256) {
    float zv = zxbcdt[(size_t)t * NPROJ + d];
    float g = zv / (1.f + __expf(-zv));
    float v = y[(size_t)t * DINNER + d] * g;
    yn[(size_t)t * DINNER + d] = v;
    ss += v * v;
  }
  red[threadIdx.x] = ss;
  __syncthreads();
  for (int off = 128; off > 0; off >>= 1) {
    if (threadIdx.x < off) red[threadIdx.x] += red[threadIdx.x + off];
    __syncthreads();
  }
  float scale = rsqrtf(red[0] / (float)DINNER + 1e-5f);
  for (int d = threadIdx.x; d < DINNER; d += 256) {
    yn[(size_t)t * DINNER + d] *= scale * norm_w[d];
  }
}

// ---------------------------------------------------------------------------
extern "C" void kernel_launch(void* const* d_in, const int* in_sizes, int n_in,
                              void* d_out, int out_size, void* d_ws, size_t ws_size,
                              hipStream_t stream) {
  (void)in_sizes; (void)n_in; (void)out_size; (void)ws_size;
  const float* u       = (const float*)d_in[0];
  const float* W_in    = (const float*)d_in[1];
  const float* conv_w  = (const float*)d_in[2];
  const float* conv_b  = (const float*)d_in[3];
  const float* dt_bias = (const float*)d_in[4];
  const float* A_log   = (const float*)d_in[5];
  const float* D_param = (const float*)d_in[6];
  const float* norm_w  = (const float*)d_in[7];
  const float* W_out   = (const float*)d_in[8];
  float* out = (float*)d_out;

  // Workspace partition (floats); total ~420 MB
  float* ws = (float*)d_ws;
  size_t off = 0;
  float* zxbcdt     = ws + off; off += (size_t)L_SEQ * NPROJ;
  float* x          = ws + off; off += (size_t)L_SEQ * DINNER;
  float* Xdt        = ws + off; off += (size_t)L_SEQ * DINNER;   // reused as yn later
  float* Bm         = ws + off; off += (size_t)L_SEQ * DSTATE;
  float* Cm         = ws + off; off += (size_t)L_SEQ * DSTATE;
  float* dt         = ws + off; off += (size_t)L_SEQ * NH;
  float* dA         = ws + off; off += (size_t)L_SEQ * NH;
  float* Acum       = ws + off; off += (size_t)NH * NCHUNK * QCH;
  float* chunk_sum  = ws + off; off += (size_t)NH * NCHUNK;
  float* scores     = ws + off; off += (size_t)NCHUNK * QCH * QCH;
  float* states     = ws + off; off += (size_t)NCHUNK * NH * HD * DSTATE;
  float* prev_state = ws + off; off += (size_t)NCHUNK * NH * HD * DSTATE;
  float* y          = ws + off; off += (size_t)L_SEQ * DINNER;
  float* yn         = Xdt;  // Xdt dead after y_kernel

  // 1) in_proj: zxbcdt = u @ W_in^T   (4096 x 8512, K=2048)
  {
    int tiles = (L_SEQ / 64) * (NPROJ / 64);
    gemm_abt_wmma<<<dim3((tiles + 7) / 8, 1, 1), 256, 0, stream>>>(
        u, W_in, zxbcdt, L_SEQ, NPROJ, DMODEL, 0, 0, 0);
  }
  // 2) dt / dA
  dt_da_kernel<<<(L_SEQ * NH) / 256, 256, 0, stream>>>(zxbcdt, dt_bias, A_log, dt, dA);
  // 3) conv + SiLU + split + X*dt
  conv_silu_kernel<<<(int)(((long long)L_SEQ * CONVDIM) / 256), 256, 0, stream>>>(
      zxbcdt, conv_w, conv_b, dt, x, Xdt, Bm, Cm);
  // 4) per-(h,c) cumsum of dA
  cumsum_kernel<<<NH * NCHUNK, 256, 0, stream>>>(dA, Acum, chunk_sum);
  // 5) per-chunk scores = C @ B^T (batched over 16 chunks)
  gemm_abt_wmma<<<dim3(2, 1, NCHUNK), 256, 0, stream>>>(
      Cm, Bm, scores, QCH, QCH, DSTATE,
      (long long)QCH * DSTATE, (long long)QCH * DSTATE, (long long)QCH * QCH);
  // 6) per-chunk end states
  states_kernel<<<NH * NCHUNK, 256, 0, stream>>>(Xdt, Bm, Acum, chunk_sum, states);
  // 7) inter-chunk recurrence
  chunkscan_kernel<<<(NH * HD * DSTATE) / 256, 256, 0, stream>>>(states, chunk_sum, prev_state);
  // 8) Y = diag + off + skip
  y_kernel<<<NH * NCHUNK, 256, 0, stream>>>(Xdt, Cm, scores, Acum, prev_state, x, D_param, y);
  // 9) gate + RMSNorm
  gate_norm_kernel<<<L_SEQ, 256, 0, stream>>>(y, zxbcdt, norm_w, yn);
  // 10) out_proj: out = yn @ W_out^T  (4096 x 2048, K=4096)
  {
    int tiles = (L_SEQ / 64) * (DMODEL / 64);
    gemm_abt_wmma<<<dim3((tiles + 7) / 8, 1, 1), 256, 0, stream>>>(
        yn, W_out, out, L_SEQ, DMODEL, DINNER, 0, 0, 0);
  }
}